// GCNEncoder_26061861552478
// MI455X (gfx1250) — compile-verified
//
#include <hip/hip_runtime.h>
#include <hip/hip_bf16.h>

typedef float v2f __attribute__((ext_vector_type(2)));
typedef float v8f __attribute__((ext_vector_type(8)));

// ---------------------------------------------------------------------------
// helpers
// ---------------------------------------------------------------------------
__device__ __forceinline__ void atomic_add_f32(float* p, float v) {
  // Lowers to global_atomic_add_f32 (non-returning) on gfx1250.
  __hip_atomic_fetch_add(p, v, __ATOMIC_RELAXED, __HIP_MEMORY_SCOPE_AGENT);
}

// ---------------------------------------------------------------------------
// fill
// ---------------------------------------------------------------------------
__global__ void fill_kernel(float* __restrict__ p, float v, long long n) {
  long long i = (long long)blockIdx.x * blockDim.x + threadIdx.x;
  if (i < n) p[i] = v;
}

// ---------------------------------------------------------------------------
// degree accumulation: deg[dst[e]] += 1  (deg pre-initialized to 1.0 = self loop)
// ---------------------------------------------------------------------------
__global__ void deg_kernel(const int* __restrict__ dst, float* __restrict__ deg,
                           long long E) {
  long long e = (long long)blockIdx.x * blockDim.x + threadIdx.x;
  if (e < E) atomic_add_f32(&deg[dst[e]], 1.0f);
}

// deg -> dinv = rsqrt(max(deg, 1e-12)) in place
__global__ void dinv_kernel(float* __restrict__ deg, int n) {
  int i = blockIdx.x * blockDim.x + threadIdx.x;
  if (i < n) deg[i] = __frsqrt_rn(fmaxf(deg[i], 1e-12f));
}

// ---------------------------------------------------------------------------
// WMMA GEMM: C[nrows, FOUT] = A[nrows, 128] x W[128, FOUT]   (f32, exact path)
// Block = (FOUT/16) waves; each wave -> one 16x16 tile via 32x V_WMMA_F32_16X16X4_F32.
// A tile (16 rows x 128 = full K) staged contiguously in LDS.
// ---------------------------------------------------------------------------
template <int FOUT>
__global__ __launch_bounds__(FOUT * 2) void gemm_wmma_kernel(
    const float* __restrict__ A, const float* __restrict__ W,
    float* __restrict__ C, int nrows) {
  __shared__ float lds[16 * 128];

  const int row0 = blockIdx.x * 16;
  if (row0 >= nrows) return;
  const int tid = threadIdx.x;
  constexpr int NTHREADS = FOUT * 2;

  // Cooperative stage of the 16x128 A tile (rows are contiguous, K == 128).
  if (row0 + 16 <= nrows) {
    const float4* srcv = (const float4*)(A + (size_t)row0 * 128);
    float4* dstv = (float4*)lds;
    #pragma unroll 2
    for (int i = tid; i < 512; i += NTHREADS) dstv[i] = srcv[i];
  } else {
    for (int i = tid; i < 2048; i += NTHREADS) {
      int r = i >> 7;
      lds[i] = (row0 + r < nrows) ? A[(size_t)(row0 + r) * 128 + (i & 127)] : 0.0f;
    }
  }
  __syncthreads();

  const int lane = tid & 31;
  const int wave = tid >> 5;
  const int colbase = wave * 16;   // 16-col tile of the output
  const int half = lane >> 4;      // 0: K+{0,1}, 1: K+{2,3}
  const int m = lane & 15;

  v8f acc = {};
  #pragma unroll
  for (int k = 0; k < 128; k += 4) {
    const int kb = k + half * 2;
    // A fragment: lane holds A[m][kb], A[m][kb+1] (8B-aligned LDS read)
    v2f a = *(const v2f*)&lds[m * 128 + kb];
    // B fragment: lane holds W[kb][colbase+m], W[kb+1][colbase+m]
    v2f b;
    b.x = W[(size_t)kb * FOUT + colbase + m];
    b.y = W[(size_t)(kb + 1) * FOUT + colbase + m];
    acc = __builtin_amdgcn_wmma_f32_16x16x4_f32(false, a, false, b, (short)0,
                                                acc, false, false);
  }

  // D layout: VGPR i -> row = row0 + i + half*8, col = colbase + m
  #pragma unroll
  for (int i = 0; i < 8; ++i) {
    int row = row0 + i + half * 8;
    if (row < nrows) C[(size_t)row * FOUT + colbase + m] = acc[i];
  }
}

// ---------------------------------------------------------------------------
// Edge scatter-add:  agg[dst] += dinv[src]*dinv[dst] * h[src]
// One thread handles one float4 chunk of one edge. F/4 chunks per edge.
// ---------------------------------------------------------------------------
template <int F>
__global__ void edge_agg_kernel(const int* __restrict__ src,
                                const int* __restrict__ dst,
                                const float* __restrict__ dinv,
                                const float* __restrict__ h,
                                float* __restrict__ agg, long long E) {
  constexpr int CH = F / 4;  // chunks per edge (32 or 16) -> power of two
  long long t = (long long)blockIdx.x * blockDim.x + threadIdx.x;
  long long e = t / CH;
  int c = (int)(t & (CH - 1));
  if (e >= E) return;
  int u = src[e];
  int v = dst[e];
  float norm = dinv[u] * dinv[v];
  float4 hv = *(const float4*)&h[(size_t)u * F + c * 4];
  float* out = &agg[(size_t)v * F + c * 4];
  atomic_add_f32(out + 0, norm * hv.x);
  atomic_add_f32(out + 1, norm * hv.y);
  atomic_add_f32(out + 2, norm * hv.z);
  atomic_add_f32(out + 3, norm * hv.w);
}

// ---------------------------------------------------------------------------
// Epilogue 1: h1 = relu(agg1 + dinv^2 * h1pre + b1), in place on agg1 (F=128)
// ---------------------------------------------------------------------------
__global__ void epilogue1_kernel(float* __restrict__ agg,
                                 const float* __restrict__ hpre,
                                 const float* __restrict__ dinv,
                                 const float* __restrict__ bias, long long n) {
  long long i = (long long)blockIdx.x * blockDim.x + threadIdx.x;
  if (i >= n) return;
  int v = (int)(i >> 7);
  int c = (int)(i & 127);
  float di = dinv[v];
  float val = agg[i] + di * di * hpre[i] + bias[c];
  agg[i] = fmaxf(val, 0.0f);
}

// ---------------------------------------------------------------------------
// Epilogue 2: out += dinv^2 * h2pre + b2  (F=64, no relu)
// ---------------------------------------------------------------------------
__global__ void epilogue2_kernel(float* __restrict__ out,
                                 const float* __restrict__ hpre,
                                 const float* __restrict__ dinv,
                                 const float* __restrict__ bias, long long n) {
  long long i = (long long)blockIdx.x * blockDim.x + threadIdx.x;
  if (i >= n) return;
  int v = (int)(i >> 6);
  int c = (int)(i & 63);
  float di = dinv[v];
  out[i] += di * di * hpre[i] + bias[c];
}

// ---------------------------------------------------------------------------
// launch
// ---------------------------------------------------------------------------
extern "C" void kernel_launch(void* const* d_in, const int* in_sizes, int n_in,
                              void* d_out, int out_size, void* d_ws,
                              size_t ws_size, hipStream_t stream) {
  const float* x  = (const float*)d_in[0];
  const int*   ei = (const int*)d_in[1];
  const float* W1 = (const float*)d_in[2];
  const float* b1 = (const float*)d_in[3];
  const float* W2 = (const float*)d_in[4];
  const float* b2 = (const float*)d_in[5];
  float* out = (float*)d_out;

  const int N = in_sizes[0] / 128;          // 100000
  const long long E = in_sizes[1] / 2;      // 1600000
  const int* src = ei;
  const int* dst = ei + E;

  float* ws = (float*)d_ws;
  float* deg   = ws;                         // N            (deg -> dinv)
  float* h1pre = ws + N;                     // N*128        (reused as h2pre N*64)
  float* agg1  = h1pre + (size_t)N * 128;    // N*128        (becomes h1)
  float* h2pre = h1pre;

  const int TB = 256;
  auto blocks = [](long long n, int tb) { return (unsigned)((n + tb - 1) / tb); };

  // init: deg = 1 (self loop), agg1 = 0, out = 0
  fill_kernel<<<blocks(N, TB), TB, 0, stream>>>(deg, 1.0f, N);
  fill_kernel<<<blocks((long long)N * 128, TB), TB, 0, stream>>>(agg1, 0.0f, (long long)N * 128);
  fill_kernel<<<blocks((long long)N * 64, TB), TB, 0, stream>>>(out, 0.0f, (long long)N * 64);

  // degrees + rsqrt
  deg_kernel<<<blocks(E, TB), TB, 0, stream>>>(dst, deg, E);
  dinv_kernel<<<blocks(N, TB), TB, 0, stream>>>(deg, N);
  const float* dinv = deg;

  // layer 1: GEMM (WMMA f32), edge scatter, epilogue (self loop + bias + relu)
  gemm_wmma_kernel<128><<<(N + 15) / 16, 256, 0, stream>>>(x, W1, h1pre, N);
  edge_agg_kernel<128><<<blocks(E * 32, TB), TB, 0, stream>>>(src, dst, dinv, h1pre, agg1, E);
  epilogue1_kernel<<<blocks((long long)N * 128, TB), TB, 0, stream>>>(agg1, h1pre, dinv, b1, (long long)N * 128);
  const float* h1 = agg1;

  // layer 2: GEMM (WMMA f32) into reused buffer, scatter into d_out, epilogue
  gemm_wmma_kernel<64><<<(N + 15) / 16, 128, 0, stream>>>(h1, W2, h2pre, N);
  edge_agg_kernel<64><<<blocks(E * 16, TB), TB, 0, stream>>>(src, dst, dinv, h2pre, out, E);
  epilogue2_kernel<<<blocks((long long)N * 64, TB), TB, 0, stream>>>(out, h2pre, dinv, b2, (long long)N * 64);
}